// KNNGrouper_65000035057785
// MI455X (gfx1250) — compile-verified
//
#include <hip/hip_runtime.h>

// Problem constants (match reference)
#define B_    8
#define N_    16384
#define G_    512     // NUM_GROUPS
#define K_    32      // GROUP_SIZE
#define C_    64      // feature channels
#define FEAT_ (3 + C_)  // 67

// Try the gfx1250 async global->LDS copy path (ASYNCcnt-tracked).
#define KNN_USE_ASYNC_LDS 1

typedef float v2f __attribute__((ext_vector_type(2)));
typedef float v8f __attribute__((ext_vector_type(8)));

// ---------------------------------------------------------------------------
// Kernel 1: precompute per-point packed vector (-2x, -2y, -2z, |x|^2)
// so the center-point score s = |x|^2 - 2 c.x is a K=4 dot product.
// ---------------------------------------------------------------------------
__global__ void prep_pts_kernel(const float* __restrict__ xyz,
                                float4* __restrict__ pt4) {
  int t = blockIdx.x * blockDim.x + threadIdx.x;
  if (t >= B_ * N_) return;
  float x = xyz[3 * t + 0];
  float y = xyz[3 * t + 1];
  float z = xyz[3 * t + 2];
  pt4[t] = make_float4(-2.0f * x, -2.0f * y, -2.0f * z, x * x + y * y + z * z);
}

// ---------------------------------------------------------------------------
// Kernel 2: deterministic farthest point sampling. One workgroup per batch,
// 1024 threads (32 wave32s); each thread owns 16 points in registers.
// ---------------------------------------------------------------------------
__global__ __launch_bounds__(1024) void fps_kernel(const float* __restrict__ xyz,
                                                   float* __restrict__ centers) {
  const int b = blockIdx.x;
  const int t = threadIdx.x;
  const float* X = xyz + (size_t)b * N_ * 3;

  __shared__ float redV[32];
  __shared__ int   redN[32];
  __shared__ float selC[3];

  float px[16], py[16], pz[16], dmin[16];
  float c0x = X[0], c0y = X[1], c0z = X[2];
#pragma unroll
  for (int i = 0; i < 16; ++i) {
    int n = t + i * 1024;
    float x = X[3 * n + 0], y = X[3 * n + 1], z = X[3 * n + 2];
    px[i] = x; py[i] = y; pz[i] = z;
    float dx = x - c0x, dy = y - c0y, dz = z - c0z;
    dmin[i] = dx * dx + dy * dy + dz * dz;
  }
  if (t == 0) {
    centers[(size_t)(b * G_) * 3 + 0] = c0x;
    centers[(size_t)(b * G_) * 3 + 1] = c0y;
    centers[(size_t)(b * G_) * 3 + 2] = c0z;
  }

  for (int s = 1; s < G_; ++s) {
    // local argmax (first-index tie break, matching jnp.argmax)
    float v = -1.0f;
    int   n = 0x7fffffff;
#pragma unroll
    for (int i = 0; i < 16; ++i) {
      int ni = t + i * 1024;
      if (dmin[i] > v || (dmin[i] == v && ni < n)) { v = dmin[i]; n = ni; }
    }
    // wave32 argmax reduce
#pragma unroll
    for (int off = 16; off > 0; off >>= 1) {
      float ov = __shfl_xor(v, off, 32);
      int   on = __shfl_xor(n, off, 32);
      if (ov > v || (ov == v && on < n)) { v = ov; n = on; }
    }
    int wave = t >> 5;
    if ((t & 31) == 0) { redV[wave] = v; redN[wave] = n; }
    __syncthreads();
    if (t == 0) {
      float bv = redV[0]; int bn = redN[0];
      for (int w = 1; w < 32; ++w)
        if (redV[w] > bv || (redV[w] == bv && redN[w] < bn)) { bv = redV[w]; bn = redN[w]; }
      float cx = X[3 * bn + 0], cy = X[3 * bn + 1], cz = X[3 * bn + 2];
      selC[0] = cx; selC[1] = cy; selC[2] = cz;
      centers[(size_t)(b * G_ + s) * 3 + 0] = cx;
      centers[(size_t)(b * G_ + s) * 3 + 1] = cy;
      centers[(size_t)(b * G_ + s) * 3 + 2] = cz;
    }
    __syncthreads();
    float cx = selC[0], cy = selC[1], cz = selC[2];
#pragma unroll
    for (int i = 0; i < 16; ++i) {
      float dx = px[i] - cx, dy = py[i] - cy, dz = pz[i] - cz;
      float d = dx * dx + dy * dy + dz * dz;
      dmin[i] = fminf(dmin[i], d);
    }
  }
}

// ---------------------------------------------------------------------------
// Wave32 bitonic helpers for streaming top-k (k == warp size == 32).
// Kept set: one (dist, idx) per lane, sorted ascending across lanes.
// Single-compare compare-exchange: keepMin folded at the call site.
// ---------------------------------------------------------------------------
__device__ __forceinline__ void ce_keep(float& d, int& i, int j, bool keepMin) {
  float od = __shfl_xor(d, j, 32);
  int   oi = __shfl_xor(i, j, 32);
  bool take = keepMin ? (od < d) : (od > d);
  if (take) { d = od; i = oi; }
}

__device__ __forceinline__ void warp_sort32_asc(float& d, int& i) {
  const int lane = threadIdx.x & 31;
#pragma unroll
  for (int k = 2; k <= 32; k <<= 1) {
    const bool asc = ((lane & k) == 0);  // k==32 -> ascending everywhere
#pragma unroll
    for (int j = k >> 1; j > 0; j >>= 1) {
      const bool lower = ((lane & j) == 0);
      ce_keep(d, i, j, lower == asc);
    }
  }
}

__device__ __forceinline__ void warp_merge32_asc(float& d, int& i) {
  const int lane = threadIdx.x & 31;
#pragma unroll
  for (int j = 16; j > 0; j >>= 1) ce_keep(d, i, j, (lane & j) == 0);
}

// ---------------------------------------------------------------------------
// Kernel 3: fused WMMA distance scoring + warp-resident top-32.
// Grid: (G/16, B). Block: 512 threads = 16 wave32s, handling 16 centers.
//   - points staged to LDS via async global->LDS b128 copies (ASYNCcnt)
//   - all 16 waves cooperatively score the 512-point LDS chunk with
//     V_WMMA_F32_16X16X4_F32 (A = centers (x,y,z,1), B = (-2x,-2y,-2z,|x|^2))
//   - wave w streams scores of center w through a ballot-gated bitonic top-32
// ---------------------------------------------------------------------------
__global__ __launch_bounds__(512) void knn_kernel(const float4* __restrict__ pt4,
                                                  const float* __restrict__ centers,
                                                  int* __restrict__ knn_idx) {
  const int b    = blockIdx.y;
  const int tile = blockIdx.x;            // 16-center tile: 0..G/16-1
  const int lane = threadIdx.x & 31;
  const int wave = threadIdx.x >> 5;      // 0..15

  __shared__ float4 pts[512];             // staged prepped points  (8 KB)
  __shared__ float  sc[16][512];          // scores[center][localN] (32 KB)

  // A fragment (16x4 f32): lanes 0-15 hold K=0/1, lanes 16-31 hold K=2/3.
  const int L = lane & 15;
  const float* Cb = centers + ((size_t)b * G_ + (size_t)tile * 16 + L) * 3;
  float cx = Cb[0], cy = Cb[1], cz = Cb[2];
  v2f afrag;
  afrag.x = (lane < 16) ? cx : cz;
  afrag.y = (lane < 16) ? cy : 1.0f;

  const float4* P = pt4 + (size_t)b * N_;

#if KNN_USE_ASYNC_LDS
  // 32-bit LDS offset of this thread's staging slot (AS3 ptrs are LDS offsets)
  unsigned ldsoff =
      (unsigned)(unsigned long long)(__attribute__((address_space(3))) float4*)
          &pts[threadIdx.x];
#endif

  float kd = 3.0e38f;   // kept distances (ascending across lanes)
  int   ki = 0;

  for (int chunk = 0; chunk < N_; chunk += 512) {
#if KNN_USE_ASYNC_LDS
    {
      unsigned long long gaddr =
          (unsigned long long)(const void*)&P[chunk + threadIdx.x];
      asm volatile("global_load_async_to_lds_b128 %0, %1, off"
                   :
                   : "v"(ldsoff), "v"(gaddr)
                   : "memory");
      asm volatile("s_wait_asynccnt 0" ::: "memory");
    }
#else
    pts[threadIdx.x] = P[chunk + threadIdx.x];
#endif
    if (chunk + 512 < N_)
      __builtin_prefetch(&P[chunk + 512 + threadIdx.x], 0, 1);
    __syncthreads();

    // wave w scores points [w*32, w*32+32) against all 16 centers (2 WMMAs)
#pragma unroll
    for (int t2 = 0; t2 < 2; ++t2) {
      const int ln0 = wave * 32 + t2 * 16;       // local point-tile base
      float4 p = pts[ln0 + (lane & 15)];
      v2f bfrag;                                  // B 4x16: mirror of A layout
      bfrag.x = (lane < 16) ? p.x : p.z;
      bfrag.y = (lane < 16) ? p.y : p.w;
      v8f acc = {};
      acc = __builtin_amdgcn_wmma_f32_16x16x4_f32(
          /*neg_a=*/false, afrag, /*neg_b=*/false, bfrag,
          /*c_mod=*/(short)0, acc, /*reuse_a=*/false, /*reuse_b=*/false);
      // C/D layout: VGPR r -> M=r (lanes<16) or M=r+8 (lanes>=16), N=lane&15
      const int mbase = (lane < 16) ? 0 : 8;
#pragma unroll
      for (int r = 0; r < 8; ++r)
        sc[mbase + r][ln0 + (lane & 15)] = acc[r];
    }
    __syncthreads();

    // wave w: stream 512 scores of center w through warp top-32.
    // Fast path: if no candidate beats the current 32nd-best, skip the sort
    // entirely (wave-uniform branch; after warm-up this is the common case).
#pragma unroll 1
    for (int step = 0; step < 16; ++step) {
      const int localN = step * 32 + lane;
      float nd = sc[wave][localN];
      const float kworst = __shfl(kd, 31, 32);
      if (__ballot(nd < kworst) == 0ull) continue;
      int ni = chunk + localN;
      warp_sort32_asc(nd, ni);           // new batch ascending
      nd = __shfl_xor(nd, 31, 32);       // reverse -> descending
      ni = __shfl_xor(ni, 31, 32);
      if (nd < kd) { kd = nd; ki = ni; } // elementwise min -> bitonic
      warp_merge32_asc(kd, ki);          // restore ascending kept set
    }
    __syncthreads();
  }

  const int g = tile * 16 + wave;
  knn_idx[((size_t)b * G_ + g) * K_ + lane] = ki;
}

// ---------------------------------------------------------------------------
// Kernel 4: gather group features: [B,G,K,67] = concat(rel_xyz, features).
// One block per (b,g,k) row, one thread per output channel.
// ---------------------------------------------------------------------------
__global__ void gather_kernel(const float* __restrict__ xyz,
                              const float* __restrict__ feats,
                              const float* __restrict__ centers,
                              const int* __restrict__ knn_idx,
                              float* __restrict__ gf) {
  const int row = blockIdx.x;            // 0 .. B*G*K-1
  const int j = threadIdx.x;
  if (j >= FEAT_) return;
  const int g = (row >> 5) & (G_ - 1);
  const int b = row >> 14;
  const int idx = knn_idx[row];
  float v;
  if (j < 3)
    v = xyz[((size_t)b * N_ + idx) * 3 + j] - centers[((size_t)b * G_ + g) * 3 + j];
  else
    v = feats[((size_t)b * N_ + idx) * C_ + (j - 3)];
  gf[(size_t)row * FEAT_ + j] = v;
}

// ---------------------------------------------------------------------------
extern "C" void kernel_launch(void* const* d_in, const int* in_sizes, int n_in,
                              void* d_out, int out_size, void* d_ws, size_t ws_size,
                              hipStream_t stream) {
  (void)in_sizes; (void)n_in; (void)out_size; (void)ws_size;

  const float* xyz   = (const float*)d_in[0];   // [B,N,3]
  const float* feats = (const float*)d_in[1];   // [B,N,64]

  float* out     = (float*)d_out;
  float* gf      = out;                                         // [B,G,K,67]
  float* centers = out + (size_t)B_ * G_ * K_ * FEAT_;          // [B,G,3]
  int*   knn     = (int*)(centers + (size_t)B_ * G_ * 3);       // [B,G,K]

  float4* pt4 = (float4*)d_ws;                                  // [B,N] float4 (2 MB)

  prep_pts_kernel<<<(B_ * N_ + 255) / 256, 256, 0, stream>>>(xyz, pt4);
  fps_kernel<<<B_, 1024, 0, stream>>>(xyz, centers);
  knn_kernel<<<dim3(G_ / 16, B_), 512, 0, stream>>>(pt4, centers, knn);
  gather_kernel<<<(unsigned)(B_ * G_ * K_), 96, 0, stream>>>(xyz, feats, centers, knn, gf);
}